// GraphSAGE_49065706389777
// MI455X (gfx1250) — compile-verified
//
#include <hip/hip_runtime.h>

// ---------------------------------------------------------------------------
// GraphSAGE (2-layer, mean aggr) + edge dot decode for MI455X (gfx1250).
//
// Memory-bound (~4 GB gather/scatter vs ~10 GFLOP GEMM); x/h/z are L2-resident
// (51/51/26 MB < 192 MB L2). GEMMs use the fp32 matrix path
// V_WMMA_F32_16X16X4_F32 (full fp32 accuracy; compute is far below the memory
// roofline so low-precision WMMA buys nothing). Weights are staged into LDS
// with the Tensor Data Mover (TENSOR_LOAD_TO_LDS + s_wait_tensorcnt) and each
// block covers 80 output rows (5 M-tiles/wave) to amortize weight traffic.
// Scatter-adds use hardware global_atomic_add_f32 via unsafeAtomicAdd.
// ---------------------------------------------------------------------------

typedef __attribute__((ext_vector_type(2))) float        v2f;
typedef __attribute__((ext_vector_type(8))) float        v8f;
typedef __attribute__((ext_vector_type(4))) unsigned int v4u;
typedef __attribute__((ext_vector_type(4))) int          v4i;
typedef __attribute__((ext_vector_type(8))) int          v8i;

#define SAGE_N_NODES 100000LL

// ------------------------------------------------- TDM 2-D tile -> LDS -----
// Builds a Tensor DMA Descriptor (D#) per CDNA5 ISA §8.3/§8.4 and issues
// TENSOR_LOAD_TO_LDS. dim0 = row length (elements), dim1 = rows, stride0 =
// row pitch (elements), 4-byte elements, whole tile = whole (sub)tensor,
// groups 2/3 zero (<=2-D). Completion tracked by TENSORcnt.
__device__ inline void tdm_load_2d(const void* gaddr, unsigned lds_off_bytes,
                                   unsigned dim0, unsigned dim1,
                                   unsigned stride0) {
  unsigned long long ga = (unsigned long long)(size_t)gaddr;
  v4u g0;
  g0[0] = 1u;                                        // count=1, user desc
  g0[1] = lds_off_bytes;                             // lds_addr (bytes)
  g0[2] = (unsigned)(ga & 0xFFFFFFFFu);              // global_addr[31:0]
  g0[3] = (unsigned)((ga >> 32) & 0x01FFFFFFu)       // global_addr[56:32]
        | (2u << 30);                                // type = 2 ("image")
  v8i g1;
  g1[0] = (int)(2u << 16);                           // data_size=2 (4 bytes)
  g1[1] = (int)((dim0 & 0xFFFFu) << 16);             // tensor_dim0[15:0]
  g1[2] = (int)(((dim0 >> 16) & 0xFFFFu)             // tensor_dim0[31:16]
        | ((dim1 & 0xFFFFu) << 16));                 // tensor_dim1[15:0]
  g1[3] = (int)(((dim1 >> 16) & 0xFFFFu)             // tensor_dim1[31:16]
        | ((dim0 & 0xFFFFu) << 16));                 // tile_dim0 = dim0
  g1[4] = (int)(dim1 & 0xFFFFu);                     // tile_dim1 = dim1, tile_dim2=0
  g1[5] = (int)stride0;                              // tensor_dim0_stride[31:0]
  g1[6] = 0;                                         // stride0[47:32], dim1_stride lo
  g1[7] = 0;                                         // dim1_stride hi
  v4i z4 = {0, 0, 0, 0};
#if __has_include(<hip/amd_detail/amd_gfx1250_TDM.h>)
  v8i z8 = {0, 0, 0, 0, 0, 0, 0, 0};                 // clang-23 6-arg form
  __builtin_amdgcn_tensor_load_to_lds(g0, g1, z4, z4, z8, 0);
#else
  __builtin_amdgcn_tensor_load_to_lds(g0, g1, z4, z4, 0);   // ROCm 7.2 5-arg
#endif
}

// ---------------------------------------------------------------- zero -----
__global__ void k_zero(float* __restrict__ p, long long n) {
  long long i = (long long)blockIdx.x * blockDim.x + threadIdx.x;
  const long long stride = (long long)gridDim.x * blockDim.x;
  for (; i < n; i += stride) p[i] = 0.0f;
}

// ------------------------------------------------------- degree counts -----
__global__ void k_count(const int* __restrict__ dst, float* __restrict__ deg,
                        long long E) {
  long long e = (long long)blockIdx.x * blockDim.x + threadIdx.x;
  if (e < E) unsafeAtomicAdd(&deg[dst[e]], 1.0f);
}

// ------------------------------------------- edge gather + scatter-add -----
// agg[dst[e], :] += feat[src[e], :], d = 128. Chunk index fast-moving:
// 8 consecutive lanes cover one contiguous 128-byte row segment.
__global__ void k_scatter128(const float* __restrict__ feat,
                             const int* __restrict__ src,
                             const int* __restrict__ dst,
                             float* __restrict__ agg, long long E) {
  long long idx = (long long)blockIdx.x * blockDim.x + threadIdx.x;
  if (idx >= E * 32) return;
  long long e = idx >> 5;
  int c = (int)(idx & 31);
  int s = src[e];
  int t = dst[e];
  float4 v = ((const float4*)(feat + (long long)s * 128))[c];
  float* base = agg + (long long)t * 128 + (long long)c * 4;
  unsafeAtomicAdd(base + 0, v.x);
  unsafeAtomicAdd(base + 1, v.y);
  unsafeAtomicAdd(base + 2, v.z);
  unsafeAtomicAdd(base + 3, v.w);
}

// ------------------------------------------------------- fused SAGE GEMM ---
// out[M, DOUT] = act( (agg/deg) @ Wl + xin @ Wr + bias ),  DIN = 128.
// Block: DOUT/16 waves, each wave = one N-tile x 5 M-tiles (block = 80 rows).
// Weight K-chunks (KC rows of Wl and Wr) staged in LDS by TDM; B-fragments
// read from LDS, A-fragments from global (L2-resident activations).
// WMMA fragment layouts per ISA §7.12.2 (32-bit A 16x4, 32-bit C/D 16x16):
//   A: lane -> row M = lane&15; half-wave 0 -> K{k0,k0+1}, half 1 -> K{k0+2,k0+3}
//   B: lane -> col N = lane&15; same K split per half-wave
//   D: VGPR r -> M = 8*(lane>>4) + r, N = lane&15
template <int DOUT, int RELU>
__global__ void k_sage_gemm_tdm(const float* __restrict__ agg,
                                const float* __restrict__ deg,
                                const float* __restrict__ xin,
                                const float* __restrict__ Wl,
                                const float* __restrict__ Wr,
                                const float* __restrict__ bias,
                                float* __restrict__ out) {
  constexpr int DIN = 128;
  constexpr int KC  = 4096 / DOUT;        // K rows per LDS chunk (32KB total)
  constexpr int MT  = 5;                  // M tiles per wave (80 rows/block)
  __shared__ float smem[2 * KC * DOUT];   // [Wl chunk | Wr chunk]

  const int lane = threadIdx.x & 31;
  const int half = lane >> 4;
  const int lm   = lane & 15;
  const int nt   = threadIdx.x >> 5;      // wave id = N tile
  const int ncol = nt * 16 + lm;
  const long long mbase = (long long)blockIdx.x * (16 * MT);

  float inv[MT];
  const float* arow[MT];
  const float* xrow[MT];
#pragma unroll
  for (int mt = 0; mt < MT; ++mt) {
    const long long r = mbase + mt * 16 + lm;
    inv[mt]  = 1.0f / fmaxf(deg[r], 1.0f);
    arow[mt] = agg + r * DIN;
    xrow[mt] = xin + r * DIN;
  }

  const v8f vzero = {0, 0, 0, 0, 0, 0, 0, 0};
  v8f acc[MT];
#pragma unroll
  for (int mt = 0; mt < MT; ++mt) acc[mt] = vzero;

  for (int kc = 0; kc < DIN; kc += KC) {
    if (threadIdx.x < 32) {               // wave 0 drives the TDM
      tdm_load_2d(Wl + (long long)kc * DOUT, 0u,               DOUT, KC, DOUT);
      tdm_load_2d(Wr + (long long)kc * DOUT, KC * DOUT * 4u,   DOUT, KC, DOUT);
      __builtin_amdgcn_s_wait_tensorcnt(0);
    }
    __syncthreads();                      // LDS chunk visible to all waves

    const float* lWl = smem;
    const float* lWr = smem + KC * DOUT;
#pragma unroll 4
    for (int k0 = 0; k0 < KC; k0 += 4) {
      const int kl = k0 + half * 2;       // chunk-local K for this half-wave
      const int kg = kc + kl;             // global K
      v2f bl; bl.x = lWl[kl * DOUT + ncol];
              bl.y = lWl[(kl + 1) * DOUT + ncol];
      v2f br; br.x = lWr[kl * DOUT + ncol];
              br.y = lWr[(kl + 1) * DOUT + ncol];
#pragma unroll
      for (int mt = 0; mt < MT; ++mt) {
        v2f a;  a.x  = arow[mt][kg] * inv[mt];
                a.y  = arow[mt][kg + 1] * inv[mt];
        acc[mt] = __builtin_amdgcn_wmma_f32_16x16x4_f32(
            false, a, false, bl, (short)0, acc[mt], false, false);
        v2f ax; ax.x = xrow[mt][kg];
                ax.y = xrow[mt][kg + 1];
        acc[mt] = __builtin_amdgcn_wmma_f32_16x16x4_f32(
            false, ax, false, br, (short)0, acc[mt], false, false);
      }
    }
    __syncthreads();                      // before next chunk overwrite
  }

  const float bv = bias[ncol];
#pragma unroll
  for (int mt = 0; mt < MT; ++mt) {
#pragma unroll
    for (int r = 0; r < 8; ++r) {
      float v = acc[mt][r] + bv;
      if (RELU) v = fmaxf(v, 0.0f);
      const long long mrow = mbase + mt * 16 + half * 8 + r;
      out[mrow * (long long)DOUT + ncol] = v;
    }
  }
}

// -------------------------------------------------- edge dot decode --------
__global__ void k_edge_dot64(const float* __restrict__ z,
                             const int* __restrict__ src,
                             const int* __restrict__ dst,
                             float* __restrict__ out, long long E) {
  long long e = (long long)blockIdx.x * blockDim.x + threadIdx.x;
  if (e >= E) return;
  const float4* a = (const float4*)(z + (long long)src[e] * 64);
  const float4* b = (const float4*)(z + (long long)dst[e] * 64);
  float s = 0.0f;
#pragma unroll
  for (int i = 0; i < 16; ++i) {
    float4 va = a[i], vb = b[i];
    s += va.x * vb.x + va.y * vb.y + va.z * vb.z + va.w * vb.w;
  }
  out[e] = s;
}

// ---------------------------------------------------------------------------
extern "C" void kernel_launch(void* const* d_in, const int* in_sizes, int n_in,
                              void* d_out, int out_size, void* d_ws, size_t ws_size,
                              hipStream_t stream) {
  const float* x   = (const float*)d_in[0];
  const int*   ei  = (const int*)  d_in[1];   // [2, E] flattened
  const float* Wl1 = (const float*)d_in[2];   // [128,128]
  const float* Wr1 = (const float*)d_in[3];   // [128,128]
  const float* b1  = (const float*)d_in[4];   // [128]
  const float* Wl2 = (const float*)d_in[5];   // [128,64]
  const float* Wr2 = (const float*)d_in[6];   // [128,64]
  const float* b2  = (const float*)d_in[7];   // [64]
  float* out = (float*)d_out;

  const long long N = SAGE_N_NODES;           // 100000 = 1250 * 80
  const long long E = (long long)in_sizes[1] / 2;
  const int* src  = ei;
  const int* dstp = ei + E;

  // ws layout (floats): deg[N] | agg1[N*128] | h[N*128] | agg2[N*128] | z[N*64]
  float* deg  = (float*)d_ws;
  float* agg1 = deg  + N;
  float* h    = agg1 + N * 128;
  float* agg2 = h    + N * 128;
  float* z    = agg2 + N * 128;

  const int ZB = 4096;

  // ---- layer 1 -----------------------------------------------------------
  k_zero<<<ZB, 256, 0, stream>>>(deg, N * 129);          // deg + agg1 contiguous
  {
    unsigned g = (unsigned)((E + 255) / 256);
    k_count<<<g, 256, 0, stream>>>(dstp, deg, E);
  }
  {
    unsigned g = (unsigned)((E * 32 + 255) / 256);
    k_scatter128<<<g, 256, 0, stream>>>(x, src, dstp, agg1, E);
  }
  // h = relu(mean1 @ Wl1 + x @ Wr1 + b1); 8 waves/block, 80 rows/block
  k_sage_gemm_tdm<128, 1><<<(unsigned)(N / 80), 256, 0, stream>>>(
      agg1, deg, x, Wl1, Wr1, b1, h);

  // ---- layer 2 -----------------------------------------------------------
  k_zero<<<ZB, 256, 0, stream>>>(agg2, N * 128);
  {
    unsigned g = (unsigned)((E * 32 + 255) / 256);
    k_scatter128<<<g, 256, 0, stream>>>(h, src, dstp, agg2, E);
  }
  // z = mean2 @ Wl2 + h @ Wr2 + b2; 4 waves/block, 80 rows/block
  k_sage_gemm_tdm<64, 0><<<(unsigned)(N / 80), 128, 0, stream>>>(
      agg2, deg, h, Wl2, Wr2, b2, z);

  // ---- decode ------------------------------------------------------------
  {
    unsigned g = (unsigned)((E + 255) / 256);
    k_edge_dot64<<<g, 256, 0, stream>>>(z, src, dstp, out, E);
  }
}